// Net_15642270892741
// MI455X (gfx1250) — compile-verified
//
#include <hip/hip_runtime.h>

// Scatter-add: out = A; out[index[e], :] += B[e, :]  (f32, D_FEAT = 64)
//
// Memory-bound problem (0.25 FLOP/byte): WMMA is inapplicable. The MI455X
// levers used instead:
//   - NT temporal hint on the 320MB one-pass B stream (protects the 25.6MB
//     accumulator working set in the 192MB L2 from streaming eviction)
//   - native non-returning global_atomic_add_f32 scope:SCOPE_DEV (L2-resident
//     read-modify-write; no CAS loop, no return latency)
//   - b128 vector loads, 16 lanes per edge row, coalesced

typedef __attribute__((ext_vector_type(4))) float v4f;

__global__ __launch_bounds__(256) void scatter_add_v4_d64(
    const int* __restrict__ idx, const v4f* __restrict__ B4,
    float* __restrict__ out, unsigned long long n4) {
  unsigned long long tid =
      (unsigned long long)blockIdx.x * blockDim.x + threadIdx.x;
  if (tid >= n4) return;

  // 64 floats per row -> 16 float4 per row
  unsigned int e  = (unsigned int)(tid >> 4);
  unsigned int f4 = (unsigned int)tid & 15u;

  int dst = idx[e];  // shared by 16 adjacent lanes (L0 hit)

  // Streamed-once data: non-temporal load (global_load_b128 th:TH_LOAD_NT)
  v4f v = __builtin_nontemporal_load(&B4[tid]);

  float* p = out + ((unsigned long long)(unsigned int)dst << 6) + (f4 << 2);

  // Non-returning f32 atomic adds, device scope, resolved in L2.
  asm volatile(
      "global_atomic_add_f32 %0, %1, off scope:SCOPE_DEV\n\t"
      "global_atomic_add_f32 %0, %2, off offset:4 scope:SCOPE_DEV\n\t"
      "global_atomic_add_f32 %0, %3, off offset:8 scope:SCOPE_DEV\n\t"
      "global_atomic_add_f32 %0, %4, off offset:12 scope:SCOPE_DEV"
      :
      : "v"(p), "v"(v.x), "v"(v.y), "v"(v.z), "v"(v.w)
      : "memory");
}

// Generic fallback for feature dims other than 64 (or not divisible by 4).
__global__ __launch_bounds__(256) void scatter_add_scalar(
    const int* __restrict__ idx, const float* __restrict__ B,
    float* __restrict__ out, unsigned long long n, unsigned int d) {
  unsigned long long tid =
      (unsigned long long)blockIdx.x * blockDim.x + threadIdx.x;
  if (tid >= n) return;
  unsigned long long e = tid / d;
  unsigned int f = (unsigned int)(tid - e * d);
  int dst = idx[e];
  float v = __builtin_nontemporal_load(&B[tid]);
  float* p = out + (unsigned long long)(unsigned int)dst * d + f;
  asm volatile("global_atomic_add_f32 %0, %1, off scope:SCOPE_DEV"
               :
               : "v"(p), "v"(v)
               : "memory");
}

extern "C" void kernel_launch(void* const* d_in, const int* in_sizes, int n_in,
                              void* d_out, int out_size, void* d_ws,
                              size_t ws_size, hipStream_t stream) {
  const int*   idx = (const int*)d_in[0];    // index: (N_EDGES,)
  const float* A   = (const float*)d_in[1];  // A: (N_NODES, D)
  const float* B   = (const float*)d_in[2];  // B: (N_EDGES, D)
  float*       out = (float*)d_out;          // (N_NODES, D)

  const long long n_edges = in_sizes[0];
  const long long b_elems = in_sizes[2];
  const int d = (int)(b_elems / n_edges);

  // out starts as a copy of A (ordered on `stream`, graph-capture safe)
  hipMemcpyAsync(d_out, A, (size_t)out_size * sizeof(float),
                 hipMemcpyDeviceToDevice, stream);

  const int block = 256;  // 8 wave32 waves
  if (d == 64) {
    unsigned long long n4 = (unsigned long long)b_elems >> 2;  // float4 count
    unsigned int grid = (unsigned int)((n4 + block - 1) / block);
    scatter_add_v4_d64<<<grid, block, 0, stream>>>(idx, (const v4f*)B, out, n4);
  } else {
    unsigned long long n = (unsigned long long)b_elems;
    unsigned int grid = (unsigned int)((n + block - 1) / block);
    scatter_add_scalar<<<grid, block, 0, stream>>>(idx, B, out, n,
                                                   (unsigned int)d);
  }
}